// Generator_66864050864303
// MI455X (gfx1250) — compile-verified
//
#include <hip/hip_runtime.h>
#include <hip/hip_bf16.h>

typedef float v2f __attribute__((ext_vector_type(2)));
typedef float v8f __attribute__((ext_vector_type(8)));

#define WMMA_F32_K4(acc_, a_, b_) \
  __builtin_amdgcn_wmma_f32_16x16x4_f32(false, (a_), false, (b_), (short)0, (acc_), false, false)

// ---------------------------------------------------------------------------
// Spectral norm: sigma = ||W v||^2 / (||W v|| + 1e-12), v = W^T u / (||W^T u|| + 1e-12)
// Single-block kernels -> deterministic tree reductions.
// ---------------------------------------------------------------------------
__global__ __launch_bounds__(256) void sn_phase1(const float* __restrict__ w,
                                                 const float* __restrict__ u,
                                                 int rows, int cols,
                                                 float* __restrict__ t,
                                                 float* __restrict__ nt2) {
  __shared__ float red[256];
  int tid = threadIdx.x;
  float local = 0.f;
  for (int j = tid; j < cols; j += 256) {
    float acc = 0.f;
    for (int i = 0; i < rows; ++i) acc += w[(size_t)i * cols + j] * u[i];
    t[j] = acc;
    local += acc * acc;
  }
  red[tid] = local;
  __syncthreads();
  for (int s = 128; s > 0; s >>= 1) {
    if (tid < s) red[tid] += red[tid + s];
    __syncthreads();
  }
  if (tid == 0) *nt2 = red[0];
}

__global__ __launch_bounds__(256) void sn_phase2(const float* __restrict__ w,
                                                 const float* __restrict__ t,
                                                 int rows, int cols,
                                                 const float* __restrict__ nt2,
                                                 float* __restrict__ r2out) {
  __shared__ float red[256];
  int tid = threadIdx.x;
  float invn = 1.f / (sqrtf(*nt2) + 1e-12f);
  float local = 0.f;
  for (int i = tid; i < rows; i += 256) {
    float acc = 0.f;
    const float* wr = w + (size_t)i * cols;
    for (int j = 0; j < cols; ++j) acc += wr[j] * t[j];
    acc *= invn;
    local += acc * acc;
  }
  red[tid] = local;
  __syncthreads();
  for (int s = 128; s > 0; s >>= 1) {
    if (tid < s) red[tid] += red[tid + s];
    __syncthreads();
  }
  if (tid == 0) *r2out = red[0];  // = ||W v||^2
}

// ---------------------------------------------------------------------------
// Generic transposed conv: y[b,o,p,q] = (1/sigma) * sum x[b,c,h,w] * w[c,o,ky,kx] + bias[o]
// where p = h*S + ky - Pd, q = w*S + kx - Pd.  act: 0 = none, 1 = tanh.
// ---------------------------------------------------------------------------
__global__ __launch_bounds__(256) void convt_kernel(const float* __restrict__ x,
                                                    const float* __restrict__ w,
                                                    const float* __restrict__ bias,
                                                    const float* __restrict__ r2sig,
                                                    float* __restrict__ y,
                                                    int Cin, int Cout, int Hin, int Hout,
                                                    int KK, int S, int Pd, int act, int total) {
  int idx = blockIdx.x * blockDim.x + threadIdx.x;
  if (idx >= total) return;
  int q = idx % Hout;
  int tmp = idx / Hout;
  int p = tmp % Hout; tmp /= Hout;
  int o = tmp % Cout;
  int b = tmp / Cout;
  float invs = 1.f;
  if (r2sig) {
    float r2 = *r2sig;                       // ||Wv||^2
    float sigma = r2 / (sqrtf(r2) + 1e-12f); // matches reference exactly
    invs = 1.f / sigma;
  }
  float acc = 0.f;
  for (int ky = 0; ky < KK; ++ky) {
    int ty = p + Pd - ky;
    if (ty < 0 || (ty % S)) continue;
    int h = ty / S;
    if (h >= Hin) continue;
    for (int kx = 0; kx < KK; ++kx) {
      int tx = q + Pd - kx;
      if (tx < 0 || (tx % S)) continue;
      int ww = tx / S;
      if (ww >= Hin) continue;
      const float* xp = x + ((size_t)(b * Cin) * Hin + h) * Hin + ww;
      const float* wp = w + (size_t)(o * KK + ky) * KK + kx;
      size_t xstep = (size_t)Hin * Hin;
      size_t wstep = (size_t)Cout * KK * KK;
      for (int c = 0; c < Cin; ++c)
        acc += xp[(size_t)c * xstep] * wp[(size_t)c * wstep];
    }
  }
  float r = acc * invs + bias[o];
  if (act == 1) r = tanhf(r);
  y[idx] = r;
}

// ---------------------------------------------------------------------------
// BatchNorm (batch stats over B,H,W per channel) + ReLU. One block per channel.
// ---------------------------------------------------------------------------
__global__ __launch_bounds__(256) void bn_relu_kernel(const float* __restrict__ y,
                                                      const float* __restrict__ g,
                                                      const float* __restrict__ be,
                                                      float* __restrict__ xo,
                                                      int Bn, int C, int HW) {
  __shared__ float rs[256], rq[256];
  int c = blockIdx.x, tid = threadIdx.x;
  int n = Bn * HW;
  float s = 0.f, q = 0.f;
  for (int t = tid; t < n; t += 256) {
    int b = t / HW, hw = t % HW;
    float v = y[((size_t)b * C + c) * HW + hw];
    s += v; q += v * v;
  }
  rs[tid] = s; rq[tid] = q;
  __syncthreads();
  for (int st = 128; st > 0; st >>= 1) {
    if (tid < st) { rs[tid] += rs[tid + st]; rq[tid] += rq[tid + st]; }
    __syncthreads();
  }
  float mean = rs[0] / (float)n;
  float var = rq[0] / (float)n - mean * mean;
  float scale = g[c] * rsqrtf(var + 1e-5f);
  float shift = be[c] - mean * scale;
  for (int t = tid; t < n; t += 256) {
    int b = t / HW, hw = t % HW;
    size_t ix = ((size_t)b * C + c) * HW + hw;
    float o = y[ix] * scale + shift;
    xo[ix] = o > 0.f ? o : 0.f;
  }
}

// ---------------------------------------------------------------------------
// WMMA f32 GEMM: C[M,N] = A[M,K] @ B[K,N]  (optionally B given as N x K, trans),
// with optional per-row bias and optional residual epilogue C = gamma*acc + R.
// One 16x16 output tile per wave, 8 waves per block, batched via blockIdx.y.
// ---------------------------------------------------------------------------
template <bool TRANSB>
__global__ __launch_bounds__(256) void wmma_gemm_f32(const float* __restrict__ A,
                                                     const float* __restrict__ Bm,
                                                     float* __restrict__ C,
                                                     int M, int N, int K,
                                                     int lda, int ldb, int ldc,
                                                     long long sA, long long sB, long long sC,
                                                     const float* __restrict__ bias,
                                                     const float* __restrict__ resid, long long sR,
                                                     const float* __restrict__ gamma) {
  int wid = threadIdx.x >> 5, lane = threadIdx.x & 31;
  int mt_n = M >> 4, nt_n = N >> 4;
  int tile = blockIdx.x * 8 + wid;
  if (tile >= mt_n * nt_n) return;   // wave-uniform: EXEC stays all-ones for WMMA
  int mt = tile % mt_n, nt = tile / mt_n;

  const float* Ab = A + (size_t)blockIdx.y * sA;
  const float* Bb = Bm + (size_t)blockIdx.y * sB;
  int lh = lane & 15;
  int koff = (lane >> 4) << 1;        // lanes 0-15 -> K+0/K+1 ; lanes 16-31 -> K+2/K+3
  int arow = mt * 16 + lh;
  int bcol = nt * 16 + lh;

  v8f acc = {0.f, 0.f, 0.f, 0.f, 0.f, 0.f, 0.f, 0.f};
  for (int ks = 0; ks < K; ks += 4) {
    v2f a, b;
    a[0] = Ab[(size_t)arow * lda + ks + koff];
    a[1] = Ab[(size_t)arow * lda + ks + koff + 1];
    if (TRANSB) {
      b[0] = Bb[(size_t)bcol * ldb + ks + koff];
      b[1] = Bb[(size_t)bcol * ldb + ks + koff + 1];
    } else {
      b[0] = Bb[(size_t)(ks + koff) * ldb + bcol];
      b[1] = Bb[(size_t)(ks + koff + 1) * ldb + bcol];
    }
    acc = WMMA_F32_K4(acc, a, b);
  }

  float gm = gamma ? gamma[0] : 0.f;
  int mbase = mt * 16 + ((lane >> 4) << 3);   // C/D: rows v (lanes 0-15) / v+8 (lanes 16-31)
  float* Cb = C + (size_t)blockIdx.y * sC;
  const float* Rb = resid ? resid + (size_t)blockIdx.y * sR : nullptr;
#pragma unroll
  for (int v = 0; v < 8; ++v) {
    int m = mbase + v;
    float r = acc[v];
    if (bias) r += bias[m];
    if (Rb) r = gm * r + Rb[(size_t)m * ldc + bcol];
    Cb[(size_t)m * ldc + bcol] = r;
  }
}

// ---------------------------------------------------------------------------
// Fused attention logits: for block (b, i-group) compute
//   F rows (16 gathered rows of wf @ x[b] + bf) with WMMA -> LDS,
//   G tiles with WMMA -> registers, contract over k in-register:
//   logits[b,i, j=a*f+c] = sum_k F[k, c*f+a] * G[k, a*f+c]
// P i-values per block, P*KD == 16.  Avoids materializing fx/gx in HBM.
// ---------------------------------------------------------------------------
template <int CDIM, int FDIM, int KD, int P>
__global__ __launch_bounds__(256) void attn_logits_kernel(const float* __restrict__ x,
                                                          const float* __restrict__ wf,
                                                          const float* __restrict__ bf,
                                                          const float* __restrict__ wg,
                                                          const float* __restrict__ bg,
                                                          float* __restrict__ logits) {
  constexpr int N = FDIM * FDIM;
  __shared__ float Flds[16 * N];   // attn1: 16 KB, attn2: 64 KB (of 320 KB WGP LDS)
  int wid = threadIdx.x >> 5, lane = threadIdx.x & 31;
  int i0 = blockIdx.x * P;
  int bb = blockIdx.y;
  const float* xb = x + (size_t)bb * CDIM * N;
  int lh = lane & 15;
  int koff = (lane >> 4) << 1;
  // A-matrix row held by this lane: gathered wf/wg row (k = r%KD, iv = r/KD)
  int r = lh;
  int iv = r / KD, k = r % KD;
  size_t wrow = (size_t)(k * N + i0 + iv);
  int rm = (lane >> 4) << 3;

  // Phase A: F projection -> LDS
  for (int jt = wid; jt < N / 16; jt += 8) {
    int col = jt * 16 + lh;
    v8f acc = {0.f, 0.f, 0.f, 0.f, 0.f, 0.f, 0.f, 0.f};
    for (int ks = 0; ks < CDIM; ks += 4) {
      v2f a, b;
      a[0] = wf[wrow * CDIM + ks + koff];
      a[1] = wf[wrow * CDIM + ks + koff + 1];
      b[0] = xb[(size_t)(ks + koff) * N + col];
      b[1] = xb[(size_t)(ks + koff + 1) * N + col];
      acc = WMMA_F32_K4(acc, a, b);
    }
#pragma unroll
    for (int v = 0; v < 8; ++v) {
      int r2 = rm + v;
      int k2 = r2 % KD, iv2 = r2 / KD;
      Flds[r2 * N + col] = acc[v] + bf[k2 * N + i0 + iv2];
    }
  }
  __syncthreads();

  // Phase B: G projection in registers + logits contraction
  for (int jt = wid; jt < N / 16; jt += 8) {
    int col = jt * 16 + lh;
    v8f acc = {0.f, 0.f, 0.f, 0.f, 0.f, 0.f, 0.f, 0.f};
    for (int ks = 0; ks < CDIM; ks += 4) {
      v2f a, b;
      a[0] = wg[wrow * CDIM + ks + koff];
      a[1] = wg[wrow * CDIM + ks + koff + 1];
      b[0] = xb[(size_t)(ks + koff) * N + col];
      b[1] = xb[(size_t)(ks + koff + 1) * N + col];
      acc = WMMA_F32_K4(acc, a, b);
    }
    int aq = col / FDIM, cq = col % FDIM;
    int jp = cq * FDIM + aq;            // transposed spatial index for F
    float s = 0.f;
#pragma unroll
    for (int v = 0; v < 8; ++v) {
      int r2 = rm + v;
      int k2 = r2 % KD, iv2 = r2 / KD;
      float gval = acc[v] + bg[k2 * N + i0 + iv2];
      s += Flds[r2 * N + jp] * gval;
    }
    if (P == 1) {
      // all 16 k-rows belong to i0: combine half-wave partials
      s += __shfl_xor(s, 16, 32);
      if (lane < 16)
        logits[((size_t)bb * N + i0) * N + col] = s;
    } else {
      // lanes 0-15 hold rows for i0, lanes 16-31 for i0+1
      int ivw = lane >> 4;
      logits[((size_t)bb * N + i0 + ivw) * N + col] = s;
    }
  }
}

// ---------------------------------------------------------------------------
// Row softmax, in place. One block per row.
// ---------------------------------------------------------------------------
__global__ __launch_bounds__(256) void softmax_rows(float* __restrict__ p, int N) {
  __shared__ float red[256];
  int tid = threadIdx.x;
  float* row = p + (size_t)blockIdx.x * N;
  float m = -3.4e38f;
  for (int j = tid; j < N; j += 256) m = fmaxf(m, row[j]);
  red[tid] = m;
  __syncthreads();
  for (int s = 128; s > 0; s >>= 1) {
    if (tid < s) red[tid] = fmaxf(red[tid], red[tid + s]);
    __syncthreads();
  }
  float mx = red[0];
  __syncthreads();
  float sum = 0.f;
  for (int j = tid; j < N; j += 256) {
    float e = expf(row[j] - mx);
    row[j] = e;
    sum += e;
  }
  red[tid] = sum;
  __syncthreads();
  for (int s = 128; s > 0; s >>= 1) {
    if (tid < s) red[tid] += red[tid + s];
    __syncthreads();
  }
  float inv = 1.f / red[0];
  for (int j = tid; j < N; j += 256) row[j] *= inv;
}

// ---------------------------------------------------------------------------
// Host orchestration
// ---------------------------------------------------------------------------
static inline int cdiv(int a, int b) { return (a + b - 1) / b; }

extern "C" void kernel_launch(void* const* d_in, const int* in_sizes, int n_in,
                              void* d_out, int out_size, void* d_ws, size_t ws_size,
                              hipStream_t stream) {
  (void)in_sizes; (void)n_in; (void)out_size; (void)ws_size;
  const float* z   = (const float*)d_in[0];
  const float* w1  = (const float*)d_in[1];  const float* b1  = (const float*)d_in[2];
  const float* u1  = (const float*)d_in[3];  const float* g1  = (const float*)d_in[4];
  const float* be1 = (const float*)d_in[5];
  const float* w2  = (const float*)d_in[6];  const float* b2  = (const float*)d_in[7];
  const float* u2  = (const float*)d_in[8];  const float* g2  = (const float*)d_in[9];
  const float* be2 = (const float*)d_in[10];
  const float* w3  = (const float*)d_in[11]; const float* b3  = (const float*)d_in[12];
  const float* u3  = (const float*)d_in[13]; const float* g3  = (const float*)d_in[14];
  const float* be3 = (const float*)d_in[15];
  const float* w4  = (const float*)d_in[16]; const float* b4  = (const float*)d_in[17];
  const float* u4  = (const float*)d_in[18]; const float* g4  = (const float*)d_in[19];
  const float* be4 = (const float*)d_in[20];
  const float* wl  = (const float*)d_in[21]; const float* bl  = (const float*)d_in[22];
  const float* wf1 = (const float*)d_in[23]; const float* bf1 = (const float*)d_in[24];
  const float* wg1 = (const float*)d_in[25]; const float* bg1 = (const float*)d_in[26];
  const float* wh1 = (const float*)d_in[27]; const float* bh1 = (const float*)d_in[28];
  const float* gm1 = (const float*)d_in[29];
  const float* wf2 = (const float*)d_in[30]; const float* bf2 = (const float*)d_in[31];
  const float* wg2 = (const float*)d_in[32]; const float* bg2 = (const float*)d_in[33];
  const float* wh2 = (const float*)d_in[34]; const float* bh2 = (const float*)d_in[35];
  const float* gm2 = (const float*)d_in[36];

  float* ws = (float*)d_ws;
  // workspace layout (floats)
  float* accv = ws;                 // [0..3]=nt2(w1..w4), [4..7]=r2(w1..w4)
  float* t1 = ws + 8;               // 8192
  float* t2 = t1 + 8192;            // 4096
  float* t3 = t2 + 4096;            // 2048
  float* t4 = t3 + 2048;            // 1024
  float* y1  = t4 + 1024;           // 4*512*16   = 32768
  float* x1  = y1 + 32768;          // 32768
  float* y2  = x1 + 32768;          // 4*256*64   = 65536
  float* x2  = y2 + 65536;          // 65536
  float* y3  = x2 + 65536;          // 4*128*256  = 131072
  float* x3  = y3 + 131072;         // 131072
  float* hx1 = x3 + 131072;         // 131072
  float* x3a = hx1 + 131072;        // 131072
  float* y4  = x3a + 131072;        // 4*64*1024  = 262144
  float* x4  = y4 + 262144;         // 262144
  float* hx2 = x4 + 262144;         // 262144
  float* x4a = hx2 + 262144;        // 262144

  float* out_img = (float*)d_out;                 // 4*3*64*64 = 49152
  float* p1 = out_img + 49152;                    // 4*256*256 = 262144
  float* p2 = p1 + 262144;                        // 4*1024*1024

  // ---- spectral norms (sigma^2-ish scalars into accv[4..7]) ----
  sn_phase1<<<1, 256, 0, stream>>>(w1, u1, 100, 8192, t1, accv + 0);
  sn_phase2<<<1, 256, 0, stream>>>(w1, t1, 100, 8192, accv + 0, accv + 4);
  sn_phase1<<<1, 256, 0, stream>>>(w2, u2, 512, 4096, t2, accv + 1);
  sn_phase2<<<1, 256, 0, stream>>>(w2, t2, 512, 4096, accv + 1, accv + 5);
  sn_phase1<<<1, 256, 0, stream>>>(w3, u3, 256, 2048, t3, accv + 2);
  sn_phase2<<<1, 256, 0, stream>>>(w3, t3, 256, 2048, accv + 2, accv + 6);
  sn_phase1<<<1, 256, 0, stream>>>(w4, u4, 128, 1024, t4, accv + 3);
  sn_phase2<<<1, 256, 0, stream>>>(w4, t4, 128, 1024, accv + 3, accv + 7);

  // ---- layer 1: convT(z) s1 p0 (1x1 -> 4x4) + BN + ReLU ----
  convt_kernel<<<cdiv(32768, 256), 256, 0, stream>>>(z, w1, b1, accv + 4, y1,
                                                     100, 512, 1, 4, 4, 1, 0, 0, 32768);
  bn_relu_kernel<<<512, 256, 0, stream>>>(y1, g1, be1, x1, 4, 512, 16);

  // ---- layer 2: 4x4 -> 8x8 ----
  convt_kernel<<<cdiv(65536, 256), 256, 0, stream>>>(x1, w2, b2, accv + 5, y2,
                                                     512, 256, 4, 8, 4, 2, 1, 0, 65536);
  bn_relu_kernel<<<256, 256, 0, stream>>>(y2, g2, be2, x2, 4, 256, 64);

  // ---- layer 3: 8x8 -> 16x16 ----
  convt_kernel<<<cdiv(131072, 256), 256, 0, stream>>>(x2, w3, b3, accv + 6, y3,
                                                      256, 128, 8, 16, 4, 2, 1, 0, 131072);
  bn_relu_kernel<<<128, 256, 0, stream>>>(y3, g3, be3, x3, 4, 128, 256);

  // ---- self-attention 1 (C=128, f=16, N=256, k=16) ----
  attn_logits_kernel<128, 16, 16, 1><<<dim3(256, 4), 256, 0, stream>>>(x3, wf1, bf1, wg1, bg1, p1);
  softmax_rows<<<4 * 256, 256, 0, stream>>>(p1, 256);
  // hx1 = wh1 @ x3 + bh1
  wmma_gemm_f32<false><<<dim3(cdiv(8 * 16, 8), 4), 256, 0, stream>>>(
      wh1, x3, hx1, 128, 256, 128, 128, 256, 256,
      0LL, 32768LL, 32768LL, bh1, nullptr, 0LL, nullptr);
  // x3a = gamma1 * (hx1 @ attn1^T) + x3
  wmma_gemm_f32<true><<<dim3(cdiv(8 * 16, 8), 4), 256, 0, stream>>>(
      hx1, p1, x3a, 128, 256, 256, 256, 256, 256,
      32768LL, 65536LL, 32768LL, nullptr, x3, 32768LL, gm1);

  // ---- layer 4: 16x16 -> 32x32 ----
  convt_kernel<<<cdiv(262144, 256), 256, 0, stream>>>(x3a, w4, b4, accv + 7, y4,
                                                      128, 64, 16, 32, 4, 2, 1, 0, 262144);
  bn_relu_kernel<<<64, 256, 0, stream>>>(y4, g4, be4, x4, 4, 64, 1024);

  // ---- self-attention 2 (C=64, f=32, N=1024, k=8; 2 i-values per block) ----
  attn_logits_kernel<64, 32, 8, 2><<<dim3(512, 4), 256, 0, stream>>>(x4, wf2, bf2, wg2, bg2, p2);
  softmax_rows<<<4 * 1024, 256, 0, stream>>>(p2, 1024);
  // hx2 = wh2 @ x4 + bh2
  wmma_gemm_f32<false><<<dim3(cdiv(4 * 64, 8), 4), 256, 0, stream>>>(
      wh2, x4, hx2, 64, 1024, 64, 64, 1024, 1024,
      0LL, 65536LL, 65536LL, bh2, nullptr, 0LL, nullptr);
  // x4a = gamma2 * (hx2 @ attn2^T) + x4
  wmma_gemm_f32<true><<<dim3(cdiv(4 * 64, 8), 4), 256, 0, stream>>>(
      hx2, p2, x4a, 64, 1024, 1024, 1024, 1024, 1024,
      65536LL, 1048576LL, 65536LL, nullptr, x4, 65536LL, gm2);

  // ---- final layer: 32x32 -> 64x64, tanh, straight into d_out ----
  convt_kernel<<<cdiv(49152, 256), 256, 0, stream>>>(x4a, wl, bl, nullptr, out_img,
                                                     64, 3, 32, 64, 4, 2, 1, 1, 49152);
}